// BNB8bitEmbedding_77068893159748
// MI455X (gfx1250) — compile-verified
//
#include <hip/hip_runtime.h>
#include <cstdint>
#include <cstddef>

// ---------------------------------------------------------------------------
// BNB 8-bit embedding gather + dequant for MI455X (gfx1250).
//
// out[tok, :] = float(q_weight[x[tok], :]) * (absmax[x[tok]] / 127)
//
// Memory-bound op (~128MB fp32 out, ~33MB int8 in -> ~7us floor @ 23.3TB/s).
// No matmul => no WMMA. The CDNA5 feature that fits is the Tensor Data Mover
// gather mode: one tensor_load_to_lds with 16 packed 16-bit row indices DMAs
// 16 indexed embedding rows (16KB) into LDS, tracked on TENSORcnt, while the
// waves dequantize the previous stage from LDS (double buffered).
// Output uses non-temporal b128 stores so the 128MB one-shot stream does not
// evict the 49MB table from the 192MB L2.
// ---------------------------------------------------------------------------

typedef unsigned int u32x4 __attribute__((ext_vector_type(4)));
typedef int          i32x4 __attribute__((ext_vector_type(4)));
typedef int          i32x8 __attribute__((ext_vector_type(8)));
typedef float        f32x4 __attribute__((ext_vector_type(4)));

#define VOCAB       50257
#define DIM         1024
#define NTOK        (8 * 4096)          // B*S = 32768 tokens
#define TPS         16                  // tokens (rows) per TDM gather stage
#define STAGE_BYTES (TPS * DIM)         // 16 KB per stage buffer
#define BLOCKS      512
#define STAGES_PB   ((NTOK / TPS) / BLOCKS)   // 2048 / 512 = 4 stages/block
#define THREADS     256

// Build a gather-mode Tensor DMA Descriptor (D#) and issue TENSOR_LOAD_TO_LDS.
// Executed by wave 0 only. All descriptor fields are wave-uniform (SGPRs).
__device__ __forceinline__ void issue_tdm_gather(const int8_t* __restrict__ qw,
                                                 const int* __restrict__ x,
                                                 int stage, uint32_t lds_addr)
{
  // 16 workgroup-uniform token ids for this stage (uniform scalar loads).
  unsigned idx[TPS];
  const int base = stage * TPS;
#pragma unroll
  for (int i = 0; i < TPS; ++i)
    idx[i] = (unsigned)__builtin_amdgcn_readfirstlane(x[base + i]);

  // ---- D# group 0 (4 dwords) ----
  // [1:0] count=1, [30] gather_index_size=0 (16-bit), [31] gather_mode=1
  // [63:32] lds_addr, [120:64] global_addr, [127:126] type=2
  const uint64_t ga = (uint64_t)(uintptr_t)qw;
  u32x4 g0;
  g0[0] = 0x80000001u;
  g0[1] = lds_addr;
  g0[2] = (uint32_t)ga;
  g0[3] = ((uint32_t)(ga >> 32) & 0x01FFFFFFu) | 0x80000000u;

  // ---- D# group 1 (8 dwords) ----
  // wg_mask=0, data_size=0 (1B), no pad/iterate/atomic-barrier.
  // tensor_dim0[79:48]=1024, tensor_dim1[111:80]=VOCAB, tile_dim0[127:112]=1024,
  // tile_dim1[143:128]=16 (#valid indices), tensor_dim0_stride[207:160]=1024.
  i32x8 g1;
  g1[0] = 0;
  g1[1] = (int)((unsigned)(DIM & 0xFFFF) << 16);        // tensor_dim0 lo16
  g1[2] = (int)(((unsigned)VOCAB & 0xFFFFu) << 16);     // dim0 hi=0 | dim1 lo16
  g1[3] = (int)((unsigned)(DIM & 0xFFFF) << 16);        // dim1 hi=0 | tile_dim0
  g1[4] = TPS;                                          // tile_dim1 | tile_dim2=0
  g1[5] = DIM;                                          // tensor_dim0_stride lo32
  g1[6] = 0;                                            // stride hi | dim1_stride lo
  g1[7] = 0;

  // ---- D# groups 2/3: 16 x 16-bit row indices ----
  i32x4 g2, g3;
#pragma unroll
  for (int i = 0; i < 4; ++i) {
    g2[i] = (int)(idx[2 * i]     | (idx[2 * i + 1] << 16));
    g3[i] = (int)(idx[8 + 2 * i] | (idx[9 + 2 * i] << 16));
  }

  // Unused 5th SGPR group (VADDR4, not used) -> zeros.
  i32x8 g4;
#pragma unroll
  for (int i = 0; i < 8; ++i) g4[i] = 0;

  // amdgpu-toolchain clang-23 6-arg form: (g0, g1, g2, g3, g4, cpol)
  __builtin_amdgcn_tensor_load_to_lds(g0, g1, g2, g3, g4, 0);
}

__global__ __launch_bounds__(THREADS)
void bnb8bit_embed_dequant_tdm(const int* __restrict__ x,
                               const int8_t* __restrict__ qw,
                               const float* __restrict__ absmax,
                               float* __restrict__ out)
{
  __shared__ __attribute__((aligned(16))) unsigned char smem[2 * STAGE_BYTES];
  // Generic->LDS: low 32 bits of the flat address are the LDS byte offset.
  const uint32_t lds_base = (uint32_t)(uintptr_t)(&smem[0]);

  const int tid = threadIdx.x;
  const int bid = blockIdx.x;

  // Prologue: wave 0 issues the gather for stage 0 into buffer 0.
  if (tid < 32)
    issue_tdm_gather(qw, x, bid, lds_base);

#pragma unroll
  for (int i = 0; i < STAGES_PB; ++i) {
    const int stage = bid + i * BLOCKS;

    if (tid < 32) {
      if (i + 1 < STAGES_PB) {
        // Prefetch next stage into the other buffer, then wait until only
        // that one is outstanding => stage i complete (per-wave in-order TDM).
        issue_tdm_gather(qw, x, bid + (i + 1) * BLOCKS,
                         lds_base + (uint32_t)(((i + 1) & 1) * STAGE_BYTES));
        __builtin_amdgcn_s_wait_tensorcnt(1);
      } else {
        __builtin_amdgcn_s_wait_tensorcnt(0);   // drain: last stage complete
      }
    }
    __syncthreads();   // stage i data visible to all 8 waves

    const unsigned char* buf = &smem[(i & 1) * STAGE_BYTES];
#pragma unroll 4
    for (int r = 0; r < TPS; ++r) {
      const int    tok   = stage * TPS + r;
      const int    row   = x[tok];                       // uniform -> s_load
      const float  scale = absmax[row] * (1.0f / 127.0f);

      // ds_load_b32: 4 int8 per thread, 256 threads cover the 1024-elem row.
      const uint32_t w =
          *reinterpret_cast<const uint32_t*>(buf + r * DIM + tid * 4);

      f32x4 f;
      f.x = (float)(int)(signed char)( w        & 0xff) * scale;
      f.y = (float)(int)(signed char)((w >> 8 ) & 0xff) * scale;
      f.z = (float)(int)(signed char)((w >> 16) & 0xff) * scale;
      f.w = (float)((int)w >> 24)                       * scale;

      // Coalesced 4KB row store, non-temporal so the 128MB output stream
      // doesn't evict the embedding table from L2.
      f32x4* dst =
          reinterpret_cast<f32x4*>(out + (size_t)tok * DIM + (size_t)tid * 4);
      __builtin_nontemporal_store(f, dst);
    }
    __syncthreads();   // all waves done with buf (i&1) before it is re-filled
  }
}

extern "C" void kernel_launch(void* const* d_in, const int* in_sizes, int n_in,
                              void* d_out, int out_size, void* d_ws, size_t ws_size,
                              hipStream_t stream)
{
  const int*    x      = (const int*)d_in[0];      // [B,S] int32 token ids
  const int8_t* qw     = (const int8_t*)d_in[1];   // [VOCAB, DIM] int8
  const float*  absmax = (const float*)d_in[2];    // [VOCAB] fp32
  float*        out    = (float*)d_out;            // [B,S,DIM] fp32

  bnb8bit_embed_dequant_tdm<<<BLOCKS, THREADS, 0, stream>>>(x, qw, absmax, out);
}